// STDWetDry_80865644249739
// MI455X (gfx1250) — compile-verified
//
#include <hip/hip_runtime.h>

typedef __attribute__((ext_vector_type(2))) float v2f;
typedef __attribute__((ext_vector_type(8))) float v8f;

// ---- tiling constants -------------------------------------------------------
// Each wave32 produces WAVE_OUT outputs from WAVE_E = 256 staged elements
// (WAVE_OUT + 31-element halo). 256 elements = 16 chunks of 16 -> one
// two-level WMMA cumsum.
constexpr int WAVE_OUT        = 224;
constexpr int WAVE_E          = 256;
constexpr int WAVES_PER_BLOCK = 8;
constexpr int BLOCK_THREADS   = WAVES_PER_BLOCK * 32;

// Reference constants
constexpr int   N_STEPS = 32;   // window
constexpr float INV_N   = 1.0f / 32.0f;
constexpr float TWO_TH  = 2.2f; // 2 * 1.1

__global__ __launch_bounds__(BLOCK_THREADS)
void stdwetdry_wmma_kernel(const float* __restrict__ x,
                           float* __restrict__ out,
                           int T)
{
    __shared__ float lds_e  [WAVES_PER_BLOCK][WAVE_E];
    __shared__ float lds_pc [WAVES_PER_BLOCK][WAVE_E];   // within-chunk prefix of x
    __shared__ float lds_pc2[WAVES_PER_BLOCK][WAVE_E];   // within-chunk prefix of x^2
    __shared__ float lds_off [WAVES_PER_BLOCK][16];      // exclusive chunk offsets (x)
    __shared__ float lds_off2[WAVES_PER_BLOCK][16];      // exclusive chunk offsets (x^2)

    const int lane = threadIdx.x & 31;
    const int wv   = threadIdx.x >> 5;

    const long long waveGlobal = (long long)blockIdx.x * WAVES_PER_BLOCK + wv;
    const long long base = waveGlobal * WAVE_OUT;   // first output index of this wave
    const long long g0   = base - 15;               // e[i] = x[g0 + i] (window t covers t-15..t+16)

    float* e   = lds_e[wv];
    float* pc  = lds_pc[wv];
    float* pc2 = lds_pc2[wv];

    // ---- Phase 1: stage 256 elements into LDS (zero-padded at boundaries) ---
    if (g0 >= 0 && g0 + WAVE_E <= (long long)T) {
        // interior fast path: fully coalesced, no predicates
        __builtin_prefetch(x + g0, 0, 0);   // global_prefetch_b8 (speculative, RT)
        #pragma unroll
        for (int q = 0; q < WAVE_E / 32; ++q) {
            int i = q * 32 + lane;
            e[i] = x[g0 + i];
        }
    } else {
        #pragma unroll
        for (int q = 0; q < WAVE_E / 32; ++q) {
            int i = q * 32 + lane;
            long long g = g0 + i;
            e[i] = (g >= 0 && g < (long long)T) ? x[g] : 0.0f;
        }
    }
    __syncthreads();

    // ---- Phase 2: two-level cumsum, level 1 via WMMA ------------------------
    // D = L16 x X  (K=16 split into 4x K=4 WMMAs, accumulate in C)
    //   A[m][k] = (k <= m)           (lower-triangular ones)
    //   B[k][n] = e[n*16 + k]        (chunk n in column n)
    //   D[m][n] = sum_{k<=m} e[n*16+k]  = inclusive prefix within chunk n
    //
    // A 16x4 f32 layout: lanes 0-15 M=0..15 hold K={0,1} in {v0,v1};
    //                    lanes 16-31 M=0..15 hold K={2,3}. B mirrors with N.
    const int half = lane >> 4;        // 0: K sub {0,1}; 1: K sub {2,3}
    const int nIdx = lane & 15;        // == M for A, == N for B/C/D

    v8f acc1 = {};   // cumsum of x
    v8f acc2 = {};   // cumsum of x^2
    #pragma unroll
    for (int ks = 0; ks < 4; ++ks) {
        const int k0 = ks * 4 + half * 2;     // global K of .x ; .y is k0+1
        v2f a, b1, b2;
        a.x = (k0     <= nIdx) ? 1.0f : 0.0f;
        a.y = (k0 + 1 <= nIdx) ? 1.0f : 0.0f;
        float bx = e[nIdx * 16 + k0];
        float by = e[nIdx * 16 + k0 + 1];
        b1.x = bx;      b1.y = by;
        b2.x = bx * bx; b2.y = by * by;
        acc1 = __builtin_amdgcn_wmma_f32_16x16x4_f32(false, a, false, b1,
                                                     (short)0, acc1, false, false);
        acc2 = __builtin_amdgcn_wmma_f32_16x16x4_f32(false, a, false, b2,
                                                     (short)0, acc2, false, false);
    }

    // Scatter D back: pc[n*16 + m] = D[m][n]
    // C/D layout: VGPR r, lanes 0-15 -> M=r ; lanes 16-31 -> M=r+8 ; N = lane%16
    #pragma unroll
    for (int r = 0; r < 8; ++r) {
        int m = r + half * 8;
        int idx = nIdx * 16 + m;
        pc [idx] = acc1[r];
        pc2[idx] = acc2[r];
    }
    __syncthreads();

    // ---- Phase 3: level-2 carry: exclusive scan of 16 chunk totals ----------
    float t1 = pc [nIdx * 16 + 15];
    float t2 = pc2[nIdx * 16 + 15];
    float s1 = t1, s2 = t2;
    #pragma unroll
    for (int d = 1; d < 16; d <<= 1) {
        float u1 = __shfl_up(s1, (unsigned)d, 32);
        float u2 = __shfl_up(s2, (unsigned)d, 32);
        if (lane >= d) { s1 += u1; s2 += u2; }
    }
    if (lane < 16) {
        lds_off [wv][lane] = s1 - t1;   // exclusive prefix of chunk totals
        lds_off2[wv][lane] = s2 - t2;
    }
    __syncthreads();

    // ---- Phase 4: windowed std -> wet/dry STE output ------------------------
    const float* off1 = lds_off[wv];
    const float* off2 = lds_off2[wv];
    #pragma unroll
    for (int q = 0; q < WAVE_OUT / 32; ++q) {
        int u = q * 32 + lane;                 // 0..223
        long long t = base + u;                // global output index
        if (t >= (long long)T) continue;
        float r;
        if (t < 15 || t >= (long long)T - 16) {
            r = 0.0f;                          // zero-padded edges (matches reference)
        } else {
            int hi = u + 31;
            float cH1 = pc [hi] + off1[hi >> 4];
            float cH2 = pc2[hi] + off2[hi >> 4];
            float cL1 = 0.0f, cL2 = 0.0f;
            if (u > 0) {
                int lo = u - 1;
                cL1 = pc [lo] + off1[lo >> 4];
                cL2 = pc2[lo] + off2[lo >> 4];
            }
            float wsum = cH1 - cL1;            // sum x over window of 32
            float wsq  = cH2 - cL2;            // sum x^2 over window
            float mean = wsum * INV_N;
            float var  = fmaxf(wsq * INV_N - mean * mean, 0.0f);
            float sigma = sqrtf(var) / TWO_TH;
            float hard  = fminf(fmaxf(rintf(sigma), 0.0f), 1.0f); // round half-even + clip
            r = (hard - sigma) + sigma;        // straight-through estimator numerics
        }
        out[t] = r;
    }
}

extern "C" void kernel_launch(void* const* d_in, const int* in_sizes, int n_in,
                              void* d_out, int out_size, void* d_ws, size_t ws_size,
                              hipStream_t stream) {
    (void)n_in; (void)d_ws; (void)ws_size;
    const float* x = (const float*)d_in[0];
    float* out = (float*)d_out;
    const int T = in_sizes[0];               // == out_size == 16777216

    const long long waves  = ((long long)T + WAVE_OUT - 1) / WAVE_OUT;
    const int       blocks = (int)((waves + WAVES_PER_BLOCK - 1) / WAVES_PER_BLOCK);

    hipLaunchKernelGGL(stdwetdry_wmma_kernel, dim3(blocks), dim3(BLOCK_THREADS),
                       0, stream, x, out, T);
}